// ConvNodeRepModule_72301479461275
// MI455X (gfx1250) — compile-verified
//
#include <hip/hip_runtime.h>
#include <hip/hip_bf16.h>

#define HID 64

typedef float v2f __attribute__((ext_vector_type(2)));
typedef float v8f __attribute__((ext_vector_type(8)));

#if defined(__HIP_DEVICE_COMPILE__)
#if !__has_builtin(__builtin_amdgcn_wmma_f32_16x16x4_f32)
#error "wmma_f32_16x16x4_f32 builtin not available on this toolchain (device pass)"
#endif
#endif

// ---------------- degree / normalization ----------------

__global__ __launch_bounds__(256) void init_deg(float* __restrict__ deg, int n) {
    int i = blockIdx.x * 256 + threadIdx.x;
    if (i < n) deg[i] = 2.0f;  // improved GCN: self-loop weight 2
}

__global__ __launch_bounds__(256) void deg_accum(const long long* __restrict__ dst,
                                                 const float* __restrict__ w,
                                                 float* __restrict__ deg, int E) {
    int e = blockIdx.x * 256 + threadIdx.x;
    if (e < E) atomicAdd(&deg[(int)dst[e]], w[e]);
}

__global__ __launch_bounds__(256) void finalize_dinv(const float* __restrict__ deg,
                                                     float* __restrict__ dinv, int n) {
    int i = blockIdx.x * 256 + threadIdx.x;
    if (i < n) {
        float d = deg[i];
        dinv[i] = (d > 0.0f) ? rsqrtf(d) : 0.0f;
    }
}

// ---------------- GEMM via full-precision f32 WMMA ----------------
// lin[N][64] = X[N][64] @ W[64][64]; one wave per 16-row tile (4 N-tiles of 16).

__global__ __launch_bounds__(256) void gemm_wmma(const float* __restrict__ X,
                                                 const float* __restrict__ W,
                                                 float* __restrict__ Y, int nrows) {
    __shared__ float Wl[HID * HID];
    int tid = threadIdx.x;
    // stage 16KB weight matrix into LDS (float4 cooperative copy)
    for (int i = tid * 4; i < HID * HID; i += 256 * 4)
        *(float4*)&Wl[i] = *(const float4*)&W[i];
    __syncthreads();

    int wave = tid >> 5;
    int lane = tid & 31;
    int tile = blockIdx.x * 8 + wave;
    if (tile * 16 >= nrows) return;          // wave-uniform: EXEC stays all-ones for WMMA
    int row0  = tile * 16;
    int m     = lane & 15;
    int khalf = (lane >> 4) * 2;             // lanes 0-15: K=k,k+1 ; lanes 16-31: K=k+2,k+3

    const float* xrow = X + (size_t)(row0 + m) * HID;
    v8f c0 = {}, c1 = {}, c2 = {}, c3 = {};

    #pragma unroll
    for (int k = 0; k < HID; k += 4) {
        int kk = k + khalf;
        v2f a;
        a.x = xrow[kk];
        a.y = xrow[kk + 1];
        const float* wr0 = &Wl[kk * HID];
        const float* wr1 = &Wl[(kk + 1) * HID];
        v2f b0, b1, b2, b3;
        b0.x = wr0[m];      b0.y = wr1[m];
        b1.x = wr0[m + 16]; b1.y = wr1[m + 16];
        b2.x = wr0[m + 32]; b2.y = wr1[m + 32];
        b3.x = wr0[m + 48]; b3.y = wr1[m + 48];
#if defined(__HIP_DEVICE_COMPILE__)
        c0 = __builtin_amdgcn_wmma_f32_16x16x4_f32(false, a, false, b0, (short)0, c0, false, false);
        c1 = __builtin_amdgcn_wmma_f32_16x16x4_f32(false, a, false, b1, (short)0, c1, false, false);
        c2 = __builtin_amdgcn_wmma_f32_16x16x4_f32(false, a, false, b2, (short)0, c2, false, false);
        c3 = __builtin_amdgcn_wmma_f32_16x16x4_f32(false, a, false, b3, (short)0, c3, false, false);
#endif
    }

    // C/D layout: VGPR r -> M=r (lanes 0-15) / M=r+8 (lanes 16-31), N = lane&15
    int mbase = (lane >> 4) * 8;
    float* yb = Y + (size_t)row0 * HID;
    #pragma unroll
    for (int r = 0; r < 8; ++r) {
        int rr = mbase + r;
        yb[rr * HID + m]      = c0[r];
        yb[rr * HID + 16 + m] = c1[r];
        yb[rr * HID + 32 + m] = c2[r];
        yb[rr * HID + 48 + m] = c3[r];
    }
}

// ---------------- aggregation ----------------

// agg[i][:] = lin[i][:] * (2 * dinv[i]^2)   (self-loop contribution, also initializes agg)
__global__ __launch_bounds__(256) void agg_self(const float* __restrict__ lin,
                                                const float* __restrict__ dinv,
                                                float* __restrict__ agg, int n) {
    int t = blockIdx.x * 256 + threadIdx.x;
    int node = t >> 4;
    if (node >= n) return;
    int fq = (t & 15) * 4;
    float s = dinv[node];
    s = 2.0f * s * s;
    float4 v = *(const float4*)&lin[(size_t)node * HID + fq];
    v.x *= s; v.y *= s; v.z *= s; v.w *= s;
    *(float4*)&agg[(size_t)node * HID + fq] = v;
}

// 32 lanes per edge, 2 features per lane -> atomic scatter-add into agg[dst]
__global__ __launch_bounds__(256) void agg_edges(const long long* __restrict__ src,
                                                 const long long* __restrict__ dst,
                                                 const float* __restrict__ w,
                                                 const float* __restrict__ dinv,
                                                 const float* __restrict__ lin,
                                                 float* __restrict__ agg, int E) {
    long long t = (long long)blockIdx.x * 256 + threadIdx.x;
    int e = (int)(t >> 5);
    if (e >= E) return;
    int lane = (int)(t & 31);
    int s = (int)src[e];
    int d = (int)dst[e];
    float norm = dinv[s] * w[e] * dinv[d];
    int f = lane * 2;
    float2 v = *(const float2*)&lin[(size_t)s * HID + f];
    float* ap = &agg[(size_t)d * HID + f];
    atomicAdd(ap,     v.x * norm);
    atomicAdd(ap + 1, v.y * norm);
}

// ---------------- bias + relu + batchnorm ----------------

__global__ void zero_stats(float* __restrict__ stats) {
    stats[threadIdx.x] = 0.0f;  // 128 threads: sum[64] + sumsq[64]
}

__global__ __launch_bounds__(256) void bias_relu_stats(float* __restrict__ agg,
                                                       const float* __restrict__ bias,
                                                       float* __restrict__ stats, int total) {
    __shared__ float s1[256];
    __shared__ float s2[256];
    int tid = threadIdx.x;
    int idx = blockIdx.x * 256 + tid;
    float v = 0.0f;
    if (idx < total) {
        int f = idx & (HID - 1);
        v = agg[idx] + bias[f];
        v = fmaxf(v, 0.0f);
        agg[idx] = v;  // in-place ReLU(agg + b)
    }
    s1[tid] = v;
    s2[tid] = v * v;
    __syncthreads();
    if (tid < HID) {  // same feature column every HID threads
        float a = s1[tid] + s1[tid + 64] + s1[tid + 128] + s1[tid + 192];
        float q = s2[tid] + s2[tid + 64] + s2[tid + 128] + s2[tid + 192];
        atomicAdd(&stats[tid], a);
        atomicAdd(&stats[HID + tid], q);
    }
}

__global__ void bn_finalize(const float* __restrict__ stats,
                            float* __restrict__ murs, float invn) {
    int f = threadIdx.x;  // 64 threads
    float mu  = stats[f] * invn;
    float var = stats[HID + f] * invn - mu * mu;
    murs[f]       = mu;
    murs[HID + f] = rsqrtf(var + 1e-5f);
}

__global__ __launch_bounds__(256) void bn_apply(const float* __restrict__ h,
                                                const float* __restrict__ murs,
                                                const float* __restrict__ gamma,
                                                const float* __restrict__ beta,
                                                float* __restrict__ out, int total) {
    int idx = blockIdx.x * 256 + threadIdx.x;
    if (idx >= total) return;
    int f = idx & (HID - 1);
    out[idx] = (h[idx] - murs[f]) * murs[HID + f] * gamma[f] + beta[f];
}

// ---------------- host launch ----------------

extern "C" void kernel_launch(void* const* d_in, const int* in_sizes, int n_in,
                              void* d_out, int out_size, void* d_ws, size_t ws_size,
                              hipStream_t stream) {
    const float*     x0    = (const float*)d_in[0];      // [N,64]
    const long long* ei    = (const long long*)d_in[1];  // [2,E] int64
    const float*     ew    = (const float*)d_in[2];      // [E]
    const float*     Ws    = (const float*)d_in[3];      // [L,64,64]
    const float*     bs    = (const float*)d_in[4];      // [L,64]
    const float*     gs    = (const float*)d_in[5];      // [L,64]
    const float*     betas = (const float*)d_in[6];      // [L,64]
    float* out = (float*)d_out;                          // [L,N,64]

    const int N = in_sizes[0] / HID;
    const int E = in_sizes[2];
    const int L = in_sizes[3] / (HID * HID);

    const long long* src = ei;
    const long long* dst = ei + E;

    float* ws    = (float*)d_ws;
    float* deg   = ws;                          // N
    float* dinv  = deg + N;                     // N
    float* lin   = dinv + N;                    // N*64
    float* agg   = lin + (size_t)N * HID;       // N*64
    float* stats = agg + (size_t)N * HID;       // 128
    float* murs  = stats + 128;                 // 128

    int nb;
    nb = (N + 255) / 256; init_deg<<<nb, 256, 0, stream>>>(deg, N);
    nb = (E + 255) / 256; deg_accum<<<nb, 256, 0, stream>>>(dst, ew, deg, E);
    nb = (N + 255) / 256; finalize_dinv<<<nb, 256, 0, stream>>>(deg, dinv, N);

    const int total = N * HID;
    const int nbElem = (total + 255) / 256;
    for (int l = 0; l < L; ++l) {
        const float* hin = (l == 0) ? x0 : (out + (size_t)(l - 1) * total);
        int tiles = (N + 15) / 16;
        nb = (tiles + 7) / 8;
        gemm_wmma<<<nb, 256, 0, stream>>>(hin, Ws + (size_t)l * HID * HID, lin, N);
        nb = (N * 16 + 255) / 256;
        agg_self<<<nb, 256, 0, stream>>>(lin, dinv, agg, N);
        nb = (int)(((long long)E * 32 + 255) / 256);
        agg_edges<<<nb, 256, 0, stream>>>(src, dst, ew, dinv, lin, agg, E);
        zero_stats<<<1, 128, 0, stream>>>(stats);
        bias_relu_stats<<<nbElem, 256, 0, stream>>>(agg, bs + l * HID, stats, total);
        bn_finalize<<<1, 64, 0, stream>>>(stats, murs, 1.0f / (float)N);
        bn_apply<<<nbElem, 256, 0, stream>>>(agg, murs, gs + l * HID, betas + l * HID,
                                             out + (size_t)l * total, total);
    }
}